// CausalSelfAttention_49228915147272
// MI455X (gfx1250) — compile-verified
//
#include <hip/hip_runtime.h>
#include <hip/hip_bf16.h>
#include <math.h>

#define B_ 4
#define T_ 2048
#define C_ 1024
#define H_ 16
#define D_ 64

typedef float v2f __attribute__((ext_vector_type(2)));
typedef float v8f __attribute__((ext_vector_type(8)));

#ifndef __has_builtin
#define __has_builtin(x) 0
#endif
#if __has_builtin(__builtin_amdgcn_global_load_async_to_lds_b128) && \
    __has_builtin(__builtin_amdgcn_s_wait_asynccnt)
#define USE_ASYNC_LDS 1
#else
#define USE_ASYNC_LDS 0
#endif

#if USE_ASYNC_LDS
typedef int v4i __attribute__((ext_vector_type(4)));
typedef __attribute__((address_space(1))) v4i GAv4i;   // global (AS1)
typedef __attribute__((address_space(3))) v4i LAv4i;   // LDS (AS3)
__device__ __forceinline__ void async_b128(const void* g, void* l) {
  __builtin_amdgcn_global_load_async_to_lds_b128((GAv4i*)g, (LAv4i*)l, 0, 0);
}
#endif

// CDNA5 fp32 WMMA: D(16x16,f32) = A(16x4,f32) x B(4x16,f32) + C
__device__ __forceinline__ v8f wmma4(v2f a, v2f b, v8f c) {
  return __builtin_amdgcn_wmma_f32_16x16x4_f32(false, a, false, b, (short)0, c,
                                               false, false);
}

// -------------------------------------------------------------------------
// Kernel 1/4: generic GEMM  out[m][n] = sum_k A[m][k]*W[n][k] + bias[n]
// 64x64 block tile, 256 threads = 8 waves (16x32 each), double-buffered
// K-chunks staged with async global->LDS (b128) when available.
// -------------------------------------------------------------------------
__global__ __launch_bounds__(256) void gemm_bias_kernel(
    const float* __restrict__ A, const float* __restrict__ W,
    const float* __restrict__ bias, float* __restrict__ out,
    int M, int N, int Kd)
{
  __shared__ float As[2][64 * 36];   // stride 36 -> conflict-free frags
  __shared__ float Ws[2][64 * 36];
  const int tid  = threadIdx.x;
  const int lane = tid & 31;
  const int wave = tid >> 5;
  const int n16  = lane & 15;
  const int hi   = lane >> 4;
  const int kh   = hi << 1;          // K sub-pair: 0 or 2
  const int mBase = blockIdx.y * 64;
  const int nBase = blockIdx.x * 64;
  const int wm = wave >> 1;
  const int wn = wave & 1;

  v8f c0 = {};
  v8f c1 = {};

  // stage one 64x32 chunk of A and W into LDS buffer `bsel`
  auto stageAW = [&](int k0, int bsel) {
#pragma unroll
    for (int i = 0; i < 2; i++) {
      int c = tid + i * 256;               // float4 chunk 0..511
      int row = c >> 3, col = (c & 7) * 4;
#if USE_ASYNC_LDS
      async_b128(A + (size_t)(mBase + row) * Kd + k0 + col,
                 &As[bsel][row * 36 + col]);
      async_b128(W + (size_t)(nBase + row) * Kd + k0 + col,
                 &Ws[bsel][row * 36 + col]);
#else
      *(float4*)&As[bsel][row * 36 + col] =
          *(const float4*)(A + (size_t)(mBase + row) * Kd + k0 + col);
      *(float4*)&Ws[bsel][row * 36 + col] =
          *(const float4*)(W + (size_t)(nBase + row) * Kd + k0 + col);
#endif
    }
  };

  const int nchunks = Kd >> 5;             // Kd/32
  stageAW(0, 0);
  for (int kc = 0; kc < nchunks; kc++) {
    const int bsel = kc & 1;
    if (kc + 1 < nchunks) {
      stageAW((kc + 1) << 5, bsel ^ 1);
#if USE_ASYNC_LDS
      __builtin_amdgcn_s_wait_asynccnt(4); // only the prefetched chunk in flight
#endif
    } else {
#if USE_ASYNC_LDS
      __builtin_amdgcn_s_wait_asynccnt(0);
#endif
    }
    __syncthreads();                       // chunk kc visible to all waves
    const float* Ab = As[bsel];
    const float* Wb = Ws[bsel];
#pragma unroll
    for (int kk = 0; kk < 32; kk += 4) {
      v2f a, b0, b1;
      const int ai = (wm * 16 + n16) * 36 + kk + kh;
      a.x = Ab[ai];  a.y = Ab[ai + 1];
      const int b0i = (wn * 32 + n16) * 36 + kk + kh;
      b0.x = Wb[b0i]; b0.y = Wb[b0i + 1];
      const int b1i = (wn * 32 + 16 + n16) * 36 + kk + kh;
      b1.x = Wb[b1i]; b1.y = Wb[b1i + 1];
      c0 = wmma4(a, b0, c0);
      c1 = wmma4(a, b1, c1);
    }
    __syncthreads();                       // done reading before overwrite
  }

#pragma unroll
  for (int r = 0; r < 8; r++) {
    int row  = mBase + wm * 16 + r + (hi << 3);
    int col0 = nBase + wn * 32 + n16;
    int col1 = col0 + 16;
    out[(size_t)row * N + col0] = c0[r] + bias[col0];
    out[(size_t)row * N + col1] = c1[r] + bias[col1];
  }
}

// -------------------------------------------------------------------------
// Kernel 2: split qkv, value mix, RMS norm q/k, rotary. One wave per (b,h,t);
// lane owns elements d=lane and d=lane+32 (the rotary pair).
// -------------------------------------------------------------------------
__global__ __launch_bounds__(256) void qkv_post_kernel(
    const float* __restrict__ qkv, const float* __restrict__ v1,
    const float* __restrict__ lambp,
    float* __restrict__ qr, float* __restrict__ kr, float* __restrict__ vout)
{
  const int lane = threadIdx.x & 31;
  const int wave = threadIdx.x >> 5;
  int idx = blockIdx.x * 8 + wave;     // row in [0, B*H*T)
  int b = idx >> 15;
  int rem = idx & 32767;
  int h = rem >> 11;
  int t = rem & 2047;

  const float lam = lambp[0];
  size_t hOff = ((size_t)b * T_ + t) * (3 * C_) + (size_t)h * D_;
  float q1  = qkv[hOff + lane];
  float q2  = qkv[hOff + lane + 32];
  float k1  = qkv[hOff + C_ + lane];
  float k2v = qkv[hOff + C_ + lane + 32];
  float vv1 = qkv[hOff + 2 * C_ + lane];
  float vv2 = qkv[hOff + 2 * C_ + lane + 32];

  size_t vIdx = (((size_t)b * H_ + h) * T_ + t) * D_;
  float nv1 = (1.f - lam) * vv1 + lam * v1[vIdx + lane];
  float nv2 = (1.f - lam) * vv2 + lam * v1[vIdx + lane + 32];
  vout[vIdx + lane]      = nv1;
  vout[vIdx + lane + 32] = nv2;

  float ssq = q1 * q1 + q2 * q2;
  float ssk = k1 * k1 + k2v * k2v;
#pragma unroll
  for (int off = 16; off >= 1; off >>= 1) {
    ssq += __shfl_xor(ssq, off);
    ssk += __shfl_xor(ssk, off);
  }
  float rq = rsqrtf(ssq * (1.f / 64.f) + 1e-6f);
  float rk = rsqrtf(ssk * (1.f / 64.f) + 1e-6f);
  q1 *= rq; q2 *= rq; k1 *= rk; k2v *= rk;

  float invf = __powf(10000.f, -(float)(2 * lane) * (1.f / 64.f));
  float ang  = (float)t * invf;
  float cs = cosf(ang), sn = sinf(ang);
  qr[vIdx + lane]      = q1 * cs - q2 * sn;
  qr[vIdx + lane + 32] = q1 * sn + q2 * cs;
  kr[vIdx + lane]      = k1 * cs - k2v * sn;
  kr[vIdx + lane + 32] = k1 * sn + k2v * cs;
}

// -------------------------------------------------------------------------
// Kernel 3: flash attention. One wave per 16-row Q tile of one (b,h),
// 2 waves/block. K/V tiles double-buffered; staged with per-wave async
// global->LDS b128 (s_wait_asynccnt pipeline) when available.
// Per-wave LDS: Q(16x68) 2x{K,V}(16x68) P(16x20) = 5760 floats.
// -------------------------------------------------------------------------
__global__ __launch_bounds__(64) void flash_attn_kernel(
    const float* __restrict__ Q, const float* __restrict__ Km,
    const float* __restrict__ V, float* __restrict__ Oattn)
{
  __shared__ float sh[2 * 5760];
  const int lane = threadIdx.x & 31;
  const int wave = threadIdx.x >> 5;
  const int n16  = lane & 15;
  const int hi   = lane >> 4;
  const int kh   = hi << 1;
  int task = blockIdx.x * 2 + wave;    // 8192 total
  int bh = task >> 7;
  int qt = task & 127;
  int b = bh >> 4, h = bh & 15;

  const float* Qh = Q  + (size_t)bh * T_ * D_;
  const float* Kh = Km + (size_t)bh * T_ * D_;
  const float* Vh = V  + (size_t)bh * T_ * D_;
  float* Qs  = sh + wave * 5760;
  float* Ks0 = Qs + 1088;
  float* Vs0 = Qs + 2176;
  float* Ks1 = Qs + 3264;
  float* Vs1 = Qs + 4352;
  float* Ps  = Qs + 5440;

  // stage Q tile (16x64 contiguous) -> LDS (stride 68), b128 copies
#pragma unroll
  for (int i = 0; i < 8; i++) {
    int c = i * 32 + lane;               // float4 chunk 0..255
    int row = c >> 4, col = (c & 15) * 4;
    *(float4*)&Qs[row * 68 + col] =
        *(const float4*)(Qh + (size_t)(qt * 16) * 64 + c * 4);
  }
  __builtin_amdgcn_wave_barrier();

  v2f aq[16];
#pragma unroll
  for (int kt = 0; kt < 16; kt++) {
    aq[kt].x = Qs[n16 * 68 + kt * 4 + kh];
    aq[kt].y = Qs[n16 * 68 + kt * 4 + kh + 1];
  }

  auto stageKV = [&](int jt, float* Kbuf, float* Vbuf) {
#pragma unroll
    for (int i = 0; i < 8; i++) {
      int c = i * 32 + lane;
      int row = c >> 4, col = (c & 15) * 4;
#if USE_ASYNC_LDS
      async_b128(Kh + (size_t)(jt * 16) * 64 + c * 4, &Kbuf[row * 68 + col]);
      async_b128(Vh + (size_t)(jt * 16) * 64 + c * 4, &Vbuf[row * 68 + col]);
#else
      *(float4*)&Kbuf[row * 68 + col] =
          *(const float4*)(Kh + (size_t)(jt * 16) * 64 + c * 4);
      *(float4*)&Vbuf[row * 68 + col] =
          *(const float4*)(Vh + (size_t)(jt * 16) * 64 + c * 4);
#endif
    }
  };

  v8f o0 = {}, o1 = {}, o2 = {}, o3 = {};
  float m8[8], l8[8];
#pragma unroll
  for (int r = 0; r < 8; r++) { m8[r] = -INFINITY; l8[r] = 0.f; }

  stageKV(0, Ks0, Vs0);
  for (int jt = 0; jt <= qt; jt++) {
    float* Ksb = (jt & 1) ? Ks1 : Ks0;
    float* Vsb = (jt & 1) ? Vs1 : Vs0;
    if (jt < qt) {
      stageKV(jt + 1, (jt & 1) ? Ks0 : Ks1, (jt & 1) ? Vs0 : Vs1);
#if USE_ASYNC_LDS
      __builtin_amdgcn_s_wait_asynccnt(16);  // tile jt done; jt+1 in flight
#endif
    } else {
#if USE_ASYNC_LDS
      __builtin_amdgcn_s_wait_asynccnt(0);
#endif
    }
    __builtin_amdgcn_wave_barrier();

    // S = Q * K^T  (16x16, K=64 in 16 steps of 4)
    v8f s = {};
#pragma unroll
    for (int kt = 0; kt < 16; kt++) {
      v2f bv;
      bv.x = Ksb[n16 * 68 + kt * 4 + kh];
      bv.y = Ksb[n16 * 68 + kt * 4 + kh + 1];
      s = wmma4(aq[kt], bv, s);
    }

    // scale + causal mask + online softmax
    float alpha[8];
#pragma unroll
    for (int r = 0; r < 8; r++) {
      float sv = s[r] * 0.125f;
      int qi = qt * 16 + r + (hi << 3);
      int ki = jt * 16 + n16;
      sv = (ki <= qi) ? sv : -INFINITY;
      float mx = sv;
      mx = fmaxf(mx, __shfl_xor(mx, 1));
      mx = fmaxf(mx, __shfl_xor(mx, 2));
      mx = fmaxf(mx, __shfl_xor(mx, 4));
      mx = fmaxf(mx, __shfl_xor(mx, 8));
      float mnew = fmaxf(m8[r], mx);
      float p = __expf(sv - mnew);
      float rs = p;
      rs += __shfl_xor(rs, 1);
      rs += __shfl_xor(rs, 2);
      rs += __shfl_xor(rs, 4);
      rs += __shfl_xor(rs, 8);
      alpha[r] = __expf(m8[r] - mnew);
      l8[r] = l8[r] * alpha[r] + rs;
      m8[r] = mnew;
      Ps[(r + (hi << 3)) * 20 + n16] = p;
    }
#pragma unroll
    for (int r = 0; r < 8; r++) {
      o0[r] *= alpha[r]; o1[r] *= alpha[r]; o2[r] *= alpha[r]; o3[r] *= alpha[r];
    }
    __builtin_amdgcn_wave_barrier();

    // O += P * Vtile
#pragma unroll
    for (int kt2 = 0; kt2 < 4; kt2++) {
      v2f ap;
      ap.x = Ps[n16 * 20 + kt2 * 4 + kh];
      ap.y = Ps[n16 * 20 + kt2 * 4 + kh + 1];
      v2f bv;
      bv.x = Vsb[(kt2 * 4 + kh) * 68 + n16];
      bv.y = Vsb[(kt2 * 4 + kh + 1) * 68 + n16];
      o0 = wmma4(ap, bv, o0);
      bv.x = Vsb[(kt2 * 4 + kh) * 68 + 16 + n16];
      bv.y = Vsb[(kt2 * 4 + kh + 1) * 68 + 16 + n16];
      o1 = wmma4(ap, bv, o1);
      bv.x = Vsb[(kt2 * 4 + kh) * 68 + 32 + n16];
      bv.y = Vsb[(kt2 * 4 + kh + 1) * 68 + 32 + n16];
      o2 = wmma4(ap, bv, o2);
      bv.x = Vsb[(kt2 * 4 + kh) * 68 + 48 + n16];
      bv.y = Vsb[(kt2 * 4 + kh + 1) * 68 + 48 + n16];
      o3 = wmma4(ap, bv, o3);
    }
  }

  // normalize and write attn in (B,T,H*D) layout for the projection GEMM
#pragma unroll
  for (int r = 0; r < 8; r++) {
    float rl = 1.f / l8[r];
    int row = qt * 16 + r + (hi << 3);
    size_t base = ((size_t)b * T_ + row) * C_ + (size_t)h * D_;
    Oattn[base + n16]      = o0[r] * rl;
    Oattn[base + 16 + n16] = o1[r] * rl;
    Oattn[base + 32 + n16] = o2[r] * rl;
    Oattn[base + 48 + n16] = o3[r] * rl;
  }
}

// -------------------------------------------------------------------------
extern "C" void kernel_launch(void* const* d_in, const int* in_sizes, int n_in,
                              void* d_out, int out_size, void* d_ws, size_t ws_size,
                              hipStream_t stream)
{
  const float* x      = (const float*)d_in[0];
  const float* v1     = (const float*)d_in[1];
  const float* W_qkv  = (const float*)d_in[2];
  const float* b_qkv  = (const float*)d_in[3];
  const float* W_proj = (const float*)d_in[4];
  const float* b_proj = (const float*)d_in[5];
  const float* lamb   = (const float*)d_in[6];

  float* out   = (float*)d_out;                       // (B,T,C)
  float* value = out + (size_t)B_ * T_ * C_;          // (B,H,T,D) second output
  float* ws    = (float*)d_ws;
  float* qkv   = ws;                                  // (B*T, 3C)  96 MB
  float* attn  = ws;                                  // reuses qkv region (32 MB)
  float* qrot  = ws + (size_t)B_ * T_ * 3 * C_;       // (B,H,T,D)  32 MB
  float* krot  = qrot + (size_t)B_ * H_ * T_ * D_;    // (B,H,T,D)  32 MB

  // 1) qkv = x @ W_qkv^T + b_qkv     M=8192 N=3072 K=1024
  gemm_bias_kernel<<<dim3((3 * C_) / 64, (B_ * T_) / 64), 256, 0, stream>>>(
      x, W_qkv, b_qkv, qkv, B_ * T_, 3 * C_, C_);

  // 2) value mix + rmsnorm + rotary (also writes `value` output)
  qkv_post_kernel<<<(B_ * H_ * T_) / 8, 256, 0, stream>>>(
      qkv, v1, lamb, qrot, krot, value);

  // 3) causal flash attention (reads value output written above)
  flash_attn_kernel<<<(B_ * H_ * (T_ / 16)) / 2, 64, 0, stream>>>(
      qrot, krot, value, attn);

  // 4) out = attn @ W_proj^T + b_proj   M=8192 N=1024 K=1024
  gemm_bias_kernel<<<dim3(C_ / 64, (B_ * T_) / 64), 256, 0, stream>>>(
      attn, W_proj, b_proj, out, B_ * T_, C_, C_);
}